// NRBS_46832323396036
// MI455X (gfx1250) — compile-verified
//
#include <hip/hip_runtime.h>
#include <hip/hip_bf16.h>

// ---------------- problem constants (match reference) ----------------
constexpr int   NPTS = 100000;   // N
constexpr int   NC   = 10;       // n (components)
constexpr int   MC   = 8;        // M_CLUSTERS
constexpr int   KNB  = 16;       // K neighbours
constexpr int   NB   = 4;        // B batch
constexpr float MUV  = 20.0f;

constexpr int   ROWP = 12;       // padded row stride for transposed W_dec (48B, 16B-aligned)

typedef __attribute__((ext_vector_type(2))) float v2f;
typedef __attribute__((ext_vector_type(8))) float v8f;

// ---------------- kernel 0: zero the enc accumulators ----------------
__global__ void NRBS_zero(float* __restrict__ p, int cnt) {
    int t = blockIdx.x * blockDim.x + threadIdx.x;
    if (t < cnt) p[t] = 0.0f;
}

// ---------------- kernel 1: enc = x @ W_enc.T via V_WMMA_F32_16X16X4_F32 ----
// A tile: 16x4 (rows 0..3 = batch, rest zero), B tile: 4x16 (cols 0..9 = n).
// f32 WMMA keeps exact f32 FMA semantics for the 100k-long reduction.
// A layout (ISA 7.12.2, 32-bit A 16x4): lane half h, VGPR0 -> K=2h, VGPR1 -> K=2h+1
// => per lane a = contiguous float2 at column nb + 2*half.  B mirrors (N=lane&15).
//
// Unconditional clamped-row loads + cndmask zeroing (no divergent branches in the
// hot loop), 4 independent accumulators / 4 WMMAs per iteration so the single
// loadcnt wait is amortized and loads pipeline.
__global__ void NRBS_enc_wmma(const float* __restrict__ x,
                              const float* __restrict__ We,
                              float* __restrict__ acc /*40 f32, pre-zeroed*/) {
    const int lane  = threadIdx.x & 31;
    const int wave  = blockIdx.x * (blockDim.x >> 5) + (threadIdx.x >> 5);
    const int nwave = gridDim.x * (blockDim.x >> 5);
    const int hf    = lane >> 4;     // lane half
    const int lid   = lane & 15;     // A: row M ; B: col N

    const bool aval = (lid < NB);
    const bool bval = (lid < NC);
    // clamp invalid lanes to row 0 so the unconditional loads are always in-bounds
    const float* __restrict__ arow = x  + (size_t)(aval ? lid : 0) * NPTS;
    const float* __restrict__ brow = We + (size_t)(bval ? lid : 0) * NPTS;

    v8f c0 = {}, c1 = {}, c2 = {}, c3 = {};
    for (int nb = wave * 16; nb < NPTS; nb += nwave * 16) {   // NPTS % 16 == 0
        const int col = nb + 2 * hf;
        v2f a0 = *(const v2f*)(arow + col);
        v2f a1 = *(const v2f*)(arow + col + 4);
        v2f a2 = *(const v2f*)(arow + col + 8);
        v2f a3 = *(const v2f*)(arow + col + 12);
        v2f b0 = *(const v2f*)(brow + col);
        v2f b1 = *(const v2f*)(brow + col + 4);
        v2f b2 = *(const v2f*)(brow + col + 8);
        v2f b3 = *(const v2f*)(brow + col + 12);
        if (!aval) { a0.x=0.f; a0.y=0.f; a1.x=0.f; a1.y=0.f;
                     a2.x=0.f; a2.y=0.f; a3.x=0.f; a3.y=0.f; }
        if (!bval) { b0.x=0.f; b0.y=0.f; b1.x=0.f; b1.y=0.f;
                     b2.x=0.f; b2.y=0.f; b3.x=0.f; b3.y=0.f; }
        c0 = __builtin_amdgcn_wmma_f32_16x16x4_f32(false, a0, false, b0, (short)0, c0, false, false);
        c1 = __builtin_amdgcn_wmma_f32_16x16x4_f32(false, a1, false, b1, (short)0, c1, false, false);
        c2 = __builtin_amdgcn_wmma_f32_16x16x4_f32(false, a2, false, b2, (short)0, c2, false, false);
        c3 = __builtin_amdgcn_wmma_f32_16x16x4_f32(false, a3, false, b3, (short)0, c3, false, false);
    }
    const v8f c = (c0 + c1) + (c2 + c3);
    // C layout: VGPR r, lanes 0-15 -> (M=r, Ncol=lane). Valid: M<4 (half 0, r<4), col<10.
    if (hf == 0 && lid < NC) {
#pragma unroll
        for (int r = 0; r < NB; ++r)
            atomicAdd(&acc[r * NC + lid], c[r]);
    }
}

// ---------------- kernel 2: 320 inverse denominators -------------------------
// invd[b*80 + c*8 + m] = 1 / ((((1/60 - lo)*sigmoid(enc[b]·W_bw[c*8+m]) + lo)*MU)^2)
__global__ void NRBS_bw(const float* __restrict__ enc,
                        const float* __restrict__ Wbw,
                        float* __restrict__ invd) {
    const int t = threadIdx.x;
    if (t >= NB * NC * MC) return;
    const int b = t / (NC * MC);
    const int j = t % (NC * MC);          // j = c*MC + m  (row of W_bw)
    float s = 0.0f;
#pragma unroll
    for (int c2 = 0; c2 < NC; ++c2)
        s += enc[b * NC + c2] * Wbw[j * NC + c2];
    const float sig = 1.0f / (1.0f + expf(-s));
    const float lo  = 4.0f / 60.0f / MUV;
    const float bwv = (1.0f / 60.0f - lo) * sig + lo;
    const float den = bwv * MUV;
    invd[t] = 1.0f / (den * den);
}

// ---------------- kernel T: transpose W_dec (10 x N) -> (N x 12 padded) ------
__global__ void NRBS_transpose(const float* __restrict__ Wd, float* __restrict__ wsT) {
    const int j = blockIdx.x * blockDim.x + threadIdx.x;
    if (j >= NPTS) return;
    float* __restrict__ row = wsT + (size_t)j * ROWP;
#pragma unroll
    for (int c = 0; c < NC; ++c) row[c] = Wd[(size_t)c * NPTS + j];
    row[10] = 0.0f; row[11] = 0.0f;
}

// ---------------- kernel 3: fused window / normalize / contract --------------
template <bool USE_T>
__global__ void NRBS_main(const float* __restrict__ Wd,
                          const float* __restrict__ wsT,
                          const float* __restrict__ nd,
                          const int*   __restrict__ nid,
                          const int*   __restrict__ lab,
                          const float* __restrict__ encG,
                          const float* __restrict__ invdG,
                          float* __restrict__ out) {
    __shared__ float s_enc[NB * NC];
    __shared__ float s_invd[NB * NC * MC];
    for (int t = threadIdx.x; t < NB * NC * MC; t += blockDim.x) s_invd[t] = invdG[t];
    for (int t = threadIdx.x; t < NB * NC;      t += blockDim.x) s_enc[t]  = encG[t];
    __syncthreads();

    const int i = blockIdx.x * blockDim.x + threadIdx.x;
    if (i >= NPTS) return;

    const int m = lab[i];
    float invd[NB][NC];
#pragma unroll
    for (int b = 0; b < NB; ++b)
#pragma unroll
        for (int c = 0; c < NC; ++c)
            invd[b][c] = s_invd[b * NC * MC + c * MC + m];

    int   ids[KNB];
    float d2[KNB];
    const int4*   __restrict__ idp = (const int4*)  (nid + (size_t)i * KNB);
    const float4* __restrict__ ndp = (const float4*)(nd  + (size_t)i * KNB);
#pragma unroll
    for (int q = 0; q < KNB / 4; ++q) {
        const int4   iv = idp[q];
        const float4 dv = ndp[q];
        ids[q*4+0] = iv.x; ids[q*4+1] = iv.y; ids[q*4+2] = iv.z; ids[q*4+3] = iv.w;
        d2[q*4+0] = dv.x*dv.x; d2[q*4+1] = dv.y*dv.y; d2[q*4+2] = dv.z*dv.z; d2[q*4+3] = dv.w*dv.w;
    }

    if (USE_T) {
#pragma unroll
        for (int k = 0; k < KNB; ++k)
            __builtin_prefetch(wsT + (size_t)ids[k] * ROWP, 0, 0);  // global_prefetch_b8
    }

    float wsum[NB][NC];
    float wb  [NB][NC];
#pragma unroll
    for (int b = 0; b < NB; ++b)
#pragma unroll
        for (int c = 0; c < NC; ++c) { wsum[b][c] = 0.0f; wb[b][c] = 0.0f; }

#pragma unroll
    for (int k = 0; k < KNB; ++k) {
        float row[NC];
        if (USE_T) {
            const float* __restrict__ rp = wsT + (size_t)ids[k] * ROWP;
            const float4 r0 = *(const float4*)(rp);
            const float4 r1 = *(const float4*)(rp + 4);
            const float2 r2 = *(const float2*)(rp + 8);
            row[0]=r0.x; row[1]=r0.y; row[2]=r0.z; row[3]=r0.w;
            row[4]=r1.x; row[5]=r1.y; row[6]=r1.z; row[7]=r1.w;
            row[8]=r2.x; row[9]=r2.y;
        } else {
#pragma unroll
            for (int c = 0; c < NC; ++c) row[c] = Wd[(size_t)c * NPTS + ids[k]];
        }
        const float dd = d2[k];
#pragma unroll
        for (int c = 0; c < NC; ++c) {
            const float bv = row[c];
#pragma unroll
            for (int b = 0; b < NB; ++b) {
                const float w = fmaxf(1.0f - dd * invd[b][c], 0.0f);
                wsum[b][c] += w;
                wb[b][c]   += w * bv;
            }
        }
    }

#pragma unroll
    for (int b = 0; b < NB; ++b) {
        float acc = 0.0f;
#pragma unroll
        for (int c = 0; c < NC; ++c)
            acc += s_enc[b * NC + c] * (wb[b][c] / wsum[b][c]);
        out[(size_t)b * NPTS + i] = acc;
    }
}

// ---------------- host launcher ----------------------------------------------
extern "C" void kernel_launch(void* const* d_in, const int* in_sizes, int n_in,
                              void* d_out, int out_size, void* d_ws, size_t ws_size,
                              hipStream_t stream) {
    const float* x   = (const float*)d_in[0];
    const float* We  = (const float*)d_in[1];
    const float* Wd  = (const float*)d_in[2];
    const float* Wbw = (const float*)d_in[3];
    const float* nd  = (const float*)d_in[4];
    const int*   nid = (const int*)  d_in[5];
    const int*   lab = (const int*)  d_in[6];
    float*       out = (float*)d_out;

    // workspace layout (floats): [0..39] enc acc, [64..383] invd, [512..] W_dec^T
    float* ws      = (float*)d_ws;
    float* enc_acc = ws;
    float* invd    = ws + 64;
    float* wsT     = ws + 512;
    const size_t need = (512 + (size_t)ROWP * NPTS) * sizeof(float);

    NRBS_zero<<<1, 64, 0, stream>>>(enc_acc, NB * NC);
    NRBS_enc_wmma<<<64, 256, 0, stream>>>(x, We, enc_acc);
    NRBS_bw<<<1, 320, 0, stream>>>(enc_acc, Wbw, invd);

    const int blocks = (NPTS + 255) / 256;
    if (ws_size >= need) {
        NRBS_transpose<<<blocks, 256, 0, stream>>>(Wd, wsT);
        NRBS_main<true><<<blocks, 256, 0, stream>>>(Wd, wsT, nd, nid, lab, enc_acc, invd, out);
    } else {
        NRBS_main<false><<<blocks, 256, 0, stream>>>(Wd, wsT, nd, nid, lab, enc_acc, invd, out);
    }
}